// ContrastMemory_43946105373207
// MI455X (gfx1250) — compile-verified
//
#include <hip/hip_runtime.h>
#include <hip/hip_bf16.h>

typedef __attribute__((ext_vector_type(2))) float v2f;
typedef __attribute__((ext_vector_type(8))) float v8f;
typedef int v4i __attribute__((vector_size(16)));

#define BDIM   64
#define DDIM   128
#define KDIM   4096
#define NPOS   5
#define WROWS  (KDIM + NPOS)      // 4101
#define TILE_N 128
#define LDSPAD 132                // row pad (floats) to avoid LDS bank conflicts
#define TINV   14.285714285714286f // 1 / 0.07

// generic -> addrspace casts; builtin wants v4i* in AS1 (global) / AS3 (LDS)
#define GLB_AS4(p) ((__attribute__((address_space(1))) v4i*)(size_t)(p))
#define LDS_AS4(p) ((__attribute__((address_space(3))) v4i*)(unsigned)(size_t)(p))

// async global->LDS copy of 16 bytes
__device__ __forceinline__ void async_g2l_b128(const float* g, float* l) {
#if __has_builtin(__builtin_amdgcn_global_load_async_to_lds_b128)
    __builtin_amdgcn_global_load_async_to_lds_b128(GLB_AS4(g), LDS_AS4(l), 0, 0);
#else
    asm volatile("global_load_async_to_lds_b128 %0, %1, off"
                 :: "v"((unsigned)(size_t)(l)),
                    "v"((unsigned long long)(size_t)(g))
                 : "memory");
#endif
}

__device__ __forceinline__ void wait_asynccnt0() {
#if __has_builtin(__builtin_amdgcn_s_wait_asynccnt)
    __builtin_amdgcn_s_wait_asynccnt(0);
#else
    asm volatile("s_wait_asynccnt 0x0" ::: "memory");
#endif
}

// ---------------------------------------------------------------------------
// Kernel A: qn[br][b][:] = normalize(mem_br[idx[b,0]]); also zero zsum.
// grid: 128 blocks (br*64+b), 128 threads (d)
// ---------------------------------------------------------------------------
__global__ void __launch_bounds__(128)
qn_kernel(const float* __restrict__ mem1, const float* __restrict__ mem2,
          const int* __restrict__ idx, float* __restrict__ qn,
          float* __restrict__ zsum) {
    int br = blockIdx.x >> 6;
    int b  = blockIdx.x & 63;
    int d  = threadIdx.x;
    const float* mem = br ? mem2 : mem1;
    int row = idx[b * (KDIM + 1)];
    float v = mem[(size_t)row * DDIM + d];
    float p = v * v;
    #pragma unroll
    for (int o = 16; o > 0; o >>= 1) p += __shfl_xor(p, o, 32);
    __shared__ float ws4[4];
    if ((threadIdx.x & 31) == 0) ws4[threadIdx.x >> 5] = p;
    __syncthreads();
    float tot = ws4[0] + ws4[1] + ws4[2] + ws4[3];
    qn[((size_t)br * BDIM + b) * DDIM + d] = v / sqrtf(tot);
    if (blockIdx.x == 0 && threadIdx.x < 2) zsum[threadIdx.x] = 0.0f;
}

// ---------------------------------------------------------------------------
// Kernel B: sim tile (64 x TILE_N) via V_WMMA_F32_16X16X4_F32, fused mem-row
// norms, label mask, per-tile top-5 candidates. mem tile staged with
// GLOBAL_LOAD_ASYNC_TO_LDS_B128 (ASYNCcnt).
// grid: (ntiles, 2), block: 256 threads (8 waves; wave w owns 16 columns)
// ---------------------------------------------------------------------------
__global__ void __launch_bounds__(256)
sim_topk_kernel(const float* __restrict__ mem1, const float* __restrict__ mem2,
                const float* __restrict__ qn, const int* __restrict__ labels,
                const int* __restrict__ blab,
                float* __restrict__ cand_val, int* __restrict__ cand_idx,
                int N, int ntiles) {
    __shared__ float mem_s[TILE_N * LDSPAD];   // staged mem tile; reused for sim
    __shared__ float qn_s[BDIM * LDSPAD];
    __shared__ int   bl_s[BDIM];

    const int tile = blockIdx.x;
    const int br   = blockIdx.y;
    const int n0   = tile * TILE_N;
    const int tid  = threadIdx.x;
    const float* mem = br ? mem2 : mem1;
    const float* qnp = qn + (size_t)br * BDIM * DDIM;

    // stage mem tile (128 rows x 128 cols): async global->LDS, OOB rows zeroed
    #pragma unroll
    for (int i = 0; i < 16; ++i) {
        int f  = tid + i * 256;          // float4 id 0..4095
        int r  = f >> 5;
        int c4 = (f & 31) << 2;
        int n  = n0 + r;
        if (n < N) {
            async_g2l_b128(mem + (size_t)n * DDIM + c4, &mem_s[r * LDSPAD + c4]);
        } else {
            *(float4*)(&mem_s[r * LDSPAD + c4]) = make_float4(0.f, 0.f, 0.f, 0.f);
        }
    }
    // stage qn (64 x 128)
    #pragma unroll
    for (int i = 0; i < 8; ++i) {
        int f  = tid + i * 256;          // 0..2047
        int r  = f >> 5;
        int c4 = (f & 31) << 2;
        *(float4*)(&qn_s[r * LDSPAD + c4]) = *(const float4*)(qnp + r * DDIM + c4);
    }
    if (tid < BDIM) bl_s[tid] = blab[tid];
    wait_asynccnt0();                    // my async tile stores are in LDS
    __syncthreads();                     // everyone's are

    const int w    = tid >> 5;
    const int lane = tid & 31;
    const int half = lane >> 4;          // K-half (A/B fragment layout)
    const int lcol = lane & 15;
    const int col  = (w << 4) + lcol;    // column within tile
    const int ncol = n0 + col;
    const bool valid = ncol < N;
    const int labn = valid ? labels[ncol] : -1;

    v8f acc[4] = {};                     // 4 M-tiles of 16 rows
    float np = 0.f;
    #pragma unroll 4
    for (int k4 = 0; k4 < 32; ++k4) {    // K = 128, 4 per WMMA
        int kk = (k4 << 2) + (half << 1);
        v2f bf = *(const v2f*)(&mem_s[col * LDSPAD + kk]);
        np += bf.x * bf.x + bf.y * bf.y; // fused ||m[n]||^2 partial (this K-half)
        #pragma unroll
        for (int mt = 0; mt < 4; ++mt) {
            v2f af = *(const v2f*)(&qn_s[((mt << 4) + lcol) * LDSPAD + kk]);
            acc[mt] = __builtin_amdgcn_wmma_f32_16x16x4_f32(
                false, af, false, bf, (short)0, acc[mt], false, false);
        }
    }
    float nrm2 = np + __shfl_xor(np, 16, 32);
    float rn = nrm2 > 0.f ? (1.0f / sqrtf(nrm2)) : 0.f;

    __syncthreads();                     // everyone done reading mem_s
    // write masked sim into mem_s rows 0..63 (C layout: lane half picks M 0-7/8-15)
    #pragma unroll
    for (int mt = 0; mt < 4; ++mt) {
        #pragma unroll
        for (int j = 0; j < 8; ++j) {
            int b  = (mt << 4) + (half << 3) + j;
            float v = acc[mt][j];
            v = (labn == bl_s[b]) ? v * rn : 0.0f;   // reference: sim * mask
            if (!valid) v = -__builtin_inff();        // padding never selected
            mem_s[b * LDSPAD + col] = v;
        }
    }
    __syncthreads();

    // per-tile top-5 per b (threads 0..63), tie-break: lower index
    if (tid < BDIM) {
        int b = tid;
        float bv[NPOS]; int bi[NPOS];
        #pragma unroll
        for (int j = 0; j < NPOS; ++j) { bv[j] = -__builtin_inff(); bi[j] = 0x7fffffff; }
        for (int c = 0; c < TILE_N; ++c) {
            float v = mem_s[b * LDSPAD + c];
            int   id = n0 + c;
            #pragma unroll
            for (int j = 0; j < NPOS; ++j) {
                bool better = (v > bv[j]) || (v == bv[j] && id < bi[j]);
                if (better) { float tv = bv[j]; int ti = bi[j];
                              bv[j] = v; bi[j] = id; v = tv; id = ti; }
            }
        }
        size_t base = ((size_t)(br * ntiles + tile) * BDIM + b) * NPOS;
        #pragma unroll
        for (int j = 0; j < NPOS; ++j) { cand_val[base + j] = bv[j]; cand_idx[base + j] = bi[j]; }
    }
}

// ---------------------------------------------------------------------------
// Kernel C: merge per-tile candidates -> global top-5; write top_sim outputs
// and knn row indices. grid: 128 blocks (br*64+b), 128 threads.
// ---------------------------------------------------------------------------
__global__ void __launch_bounds__(128)
topk_reduce_kernel(const float* __restrict__ cand_val, const int* __restrict__ cand_idx,
                   int* __restrict__ topidx, float* __restrict__ out, int ntiles) {
    int br = blockIdx.x >> 6;
    int b  = blockIdx.x & 63;
    int tid = threadIdx.x;
    int total = ntiles * NPOS;

    float bv[NPOS]; int bi[NPOS];
    #pragma unroll
    for (int j = 0; j < NPOS; ++j) { bv[j] = -__builtin_inff(); bi[j] = 0x7fffffff; }
    for (int e = tid; e < total; e += 128) {
        int t = e / NPOS, s = e - t * NPOS;
        size_t base = ((size_t)(br * ntiles + t) * BDIM + b) * NPOS + s;
        float v = cand_val[base];
        int  id = cand_idx[base];
        #pragma unroll
        for (int j = 0; j < NPOS; ++j) {
            bool better = (v > bv[j]) || (v == bv[j] && id < bi[j]);
            if (better) { float tv = bv[j]; int ti = bi[j];
                          bv[j] = v; bi[j] = id; v = tv; id = ti; }
        }
    }
    __shared__ float sv[128 * NPOS];
    __shared__ int   si[128 * NPOS];
    #pragma unroll
    for (int j = 0; j < NPOS; ++j) { sv[tid * NPOS + j] = bv[j]; si[tid * NPOS + j] = bi[j]; }
    __syncthreads();
    if (tid == 0) {
        float fv[NPOS]; int fi[NPOS];
        #pragma unroll
        for (int j = 0; j < NPOS; ++j) { fv[j] = -__builtin_inff(); fi[j] = 0x7fffffff; }
        for (int e = 0; e < 128 * NPOS; ++e) {
            float v = sv[e]; int id = si[e];
            #pragma unroll
            for (int j = 0; j < NPOS; ++j) {
                bool better = (v > fv[j]) || (v == fv[j] && id < fi[j]);
                if (better) { float tv = fv[j]; int ti = fi[j];
                              fv[j] = v; fi[j] = id; v = tv; id = ti; }
            }
        }
        // outputs: [out_v1 | out_v2 | v1_sim(br=0) | v2_sim(br=1)]
        size_t simoff = (size_t)2 * BDIM * WROWS + (size_t)br * BDIM * NPOS + (size_t)b * NPOS;
        #pragma unroll
        for (int j = 0; j < NPOS; ++j) {
            out[simoff + j] = fv[j];
            topidx[((size_t)br * BDIM + b) * NPOS + j] = fi[j];
        }
    }
}

// ---------------------------------------------------------------------------
// Kernel D: out_num = exp(w_full . other_v / T); accumulate per-branch sum.
// wave-per-row; 8 waves/block. rows = 2*64*4101.
// ---------------------------------------------------------------------------
__global__ void __launch_bounds__(256)
out_kernel(const float* __restrict__ mem1, const float* __restrict__ mem2,
           const float* __restrict__ v1, const float* __restrict__ v2,
           const int* __restrict__ idx, const int* __restrict__ topidx,
           float* __restrict__ out, float* __restrict__ zsum, int nrows) {
    __shared__ float lsum[2];
    int tid = threadIdx.x;
    if (tid < 2) lsum[tid] = 0.f;
    __syncthreads();
    int wid = tid >> 5, lane = tid & 31;
    long long r = (long long)blockIdx.x * 8 + wid;
    if (r < nrows) {
        int br  = (int)(r / ((long long)BDIM * WROWS));
        int rem = (int)(r % ((long long)BDIM * WROWS));
        int b = rem / WROWS;
        int k = rem - b * WROWS;
        const float* mem = br ? mem2 : mem1;
        const float* ov  = (br ? v1 : v2) + (size_t)b * DDIM;  // br0: memory_v1 with v2
        int row = (k < NPOS) ? topidx[((size_t)br * BDIM + b) * NPOS + k]
                             : idx[b * (KDIM + 1) + (k - NPOS + 1)];
        float4 mv = ((const float4*)(mem + (size_t)row * DDIM))[lane];
        float4 vv = ((const float4*)ov)[lane];
        float p = mv.x * vv.x + mv.y * vv.y + mv.z * vv.z + mv.w * vv.w;
        #pragma unroll
        for (int o = 16; o > 0; o >>= 1) p += __shfl_xor(p, o, 32);
        if (lane == 0) {
            float e = expf(p * TINV);
            // br=0 -> out_v2 region (offset B*WROWS); br=1 -> out_v1 region (offset 0)
            size_t off = (br == 0 ? (size_t)BDIM * WROWS : (size_t)0) + (size_t)b * WROWS + k;
            out[off] = e;
            atomicAdd(&lsum[br], e);
        }
    }
    __syncthreads();
    if (tid < 2) atomicAdd(&zsum[tid], lsum[tid]);
}

// ---------------------------------------------------------------------------
// Kernel E: out /= Z_br, Z = (sum / (B*WROWS)) * N
// ---------------------------------------------------------------------------
__global__ void __launch_bounds__(256)
scale_kernel(float* __restrict__ out, const float* __restrict__ zsum, int N) {
    int e = blockIdx.x * 256 + threadIdx.x;
    const int tot = 2 * BDIM * WROWS;
    if (e >= tot) return;
    int br = (e < BDIM * WROWS) ? 1 : 0;   // out_v1 first (branch of memory_v2)
    float Z = (zsum[br] / (float)(BDIM * WROWS)) * (float)N;
    out[e] = out[e] / Z;
}

// ---------------------------------------------------------------------------
extern "C" void kernel_launch(void* const* d_in, const int* in_sizes, int n_in,
                              void* d_out, int out_size, void* d_ws, size_t ws_size,
                              hipStream_t stream) {
    const float* v1    = (const float*)d_in[1];
    const float* v2    = (const float*)d_in[2];
    const int*   blab  = (const int*)d_in[3];
    const int*   idx   = (const int*)d_in[5];
    const int*   labs  = (const int*)d_in[6];
    const float* mem1  = (const float*)d_in[7];
    const float* mem2  = (const float*)d_in[8];
    const int N = in_sizes[6];
    const int ntiles = (N + TILE_N - 1) / TILE_N;

    char* ws = (char*)d_ws;
    float* qn       = (float*)ws;                       ws += (size_t)2 * BDIM * DDIM * 4;
    float* cand_val = (float*)ws;                       ws += (size_t)2 * ntiles * BDIM * NPOS * 4;
    int*   cand_idx = (int*)ws;                         ws += (size_t)2 * ntiles * BDIM * NPOS * 4;
    int*   topidx   = (int*)ws;                         ws += (size_t)2 * BDIM * NPOS * 4;
    float* zsum     = (float*)ws;
    float* out      = (float*)d_out;

    qn_kernel<<<dim3(2 * BDIM), dim3(128), 0, stream>>>(mem1, mem2, idx, qn, zsum);

    sim_topk_kernel<<<dim3(ntiles, 2), dim3(256), 0, stream>>>(
        mem1, mem2, qn, labs, blab, cand_val, cand_idx, N, ntiles);

    topk_reduce_kernel<<<dim3(2 * BDIM), dim3(128), 0, stream>>>(
        cand_val, cand_idx, topidx, out, ntiles);

    const int nrows = 2 * BDIM * WROWS;
    out_kernel<<<dim3((nrows + 7) / 8), dim3(256), 0, stream>>>(
        mem1, mem2, v1, v2, idx, topidx, out, zsum, nrows);

    scale_kernel<<<dim3((nrows + 255) / 256), dim3(256), 0, stream>>>(out, zsum, N);
}